// GATAutoScaler_24996709663257
// MI455X (gfx1250) — compile-verified
//
#include <hip/hip_runtime.h>

typedef __attribute__((ext_vector_type(2))) float v2f;
typedef __attribute__((ext_vector_type(8))) float v8f;
typedef __attribute__((ext_vector_type(4))) unsigned int v4u;
typedef __attribute__((ext_vector_type(4))) int v4i;
typedef __attribute__((ext_vector_type(8))) int v8i;

static inline int cdiv_i(long long a, long long b) { return (int)((a + b - 1) / b); }

// ---------------------------------------------------------------------------
// Tensor Data Mover: 2-D tile load Global -> LDS (ISA cdna5 §8 descriptor).
//   ldsOff       : byte offset of destination inside LDS
//   gptr         : global address of tile start
//   tensorW/H    : tensor extent (elements) for OOB clamping (>= tile)
//   rowStrideElt : elements between consecutive tensor rows
//   tileW/H      : tile extent (elements)
//   padICode     : pad_interval code (2<<code dwords of data between pads)
//   padACode     : pad_amount code  (code+1 dwords of LDS padding inserted)
// Issued once per wave (EXEC ignored by TDM); caller guards to one wave.
// Completion tracked with TENSORcnt.
// ---------------------------------------------------------------------------
__device__ __forceinline__ void tdm_load_2d(unsigned ldsOff, const void* gptr,
                                            unsigned tensorW, unsigned tensorH,
                                            unsigned long long rowStrideElt,
                                            unsigned tileW, unsigned tileH,
                                            unsigned padICode, unsigned padACode) {
  const unsigned long long ga = (unsigned long long)(uintptr_t)gptr;
  v4u g0;
  g0.x = 1u;                                          // count=1, user descriptor
  g0.y = ldsOff;                                      // lds_addr (bytes)
  g0.z = (unsigned)(ga & 0xffffffffu);                // global_addr[31:0]
  g0.w = (unsigned)((ga >> 32) & 0x01ffffffu)         // global_addr[56:32]
         | (2u << 30);                                // type=2 ("image")
  // group1: [17:16] data_size=2 (4B), [20] pad_enable, [24:22] pad_interval,
  //         [31:25] pad_amount, dims/strides per ISA bit layout.
  const unsigned d0 = (2u << 16) | (1u << 20) | (padICode << 22) | (padACode << 25);
  const unsigned d1 = (tensorW & 0xffffu) << 16;              // tensor_dim0[15:0]
  const unsigned d2 = (tensorW >> 16) | ((tensorH & 0xffffu) << 16);
  const unsigned d3 = (tensorH >> 16) | ((tileW & 0xffffu) << 16);
  const unsigned d4 = tileH & 0xffffu;                        // tile_dim1 (tile_dim2=0)
  const unsigned d5 = (unsigned)(rowStrideElt & 0xffffffffu); // dim0_stride[31:0]
  const unsigned d6 = (unsigned)((rowStrideElt >> 32) & 0xffffu);
  v8i g1;
  g1[0] = (int)d0; g1[1] = (int)d1; g1[2] = (int)d2; g1[3] = (int)d3;
  g1[4] = (int)d4; g1[5] = (int)d5; g1[6] = (int)d6; g1[7] = 0;
  v4i z4 = {0, 0, 0, 0};
#if defined(__clang_major__) && __clang_major__ >= 23
  v8i z8 = {0, 0, 0, 0, 0, 0, 0, 0};
  __builtin_amdgcn_tensor_load_to_lds(g0, g1, z4, z4, z8, 0);
#else
  __builtin_amdgcn_tensor_load_to_lds(g0, g1, z4, z4, 0);
#endif
}

// ---------------------------------------------------------------------------
// GEMM: C[M,Nc] = A[M,K] @ B (+bias). BT=false: B is [K,Nc]; BT=true: B is [Nc,K].
// Workgroup (8 waves) computes a 128x64 C block; each wave one 16x64 strip via
// V_WMMA_F32_16X16X4_F32. K is consumed in KC=32 chunks; A(128xKC) and
// B(KCx64 / 64xKC) panels are DMA'd to LDS by the TDM, double-buffered
// (s_wait_tensorcnt 2 keeps the next chunk's pair in flight).
// LDS row padding (via TDM pad fields): A,Bt rows 32+2 dwords; B rows 64+8.
// Requires M%16==0, Nc%64==0, K%32==0.
// ---------------------------------------------------------------------------
template <bool BT>
__global__ __launch_bounds__(256) void gemm_wmma_kernel(
    const float* __restrict__ A, const float* __restrict__ B,
    const float* __restrict__ bias, float* __restrict__ C,
    int M, int K, int Nc) {
  constexpr int KC = 32;
  constexpr int AS = 34;  // A (and Bt) LDS row stride in dwords (32 data + 2 pad)
  constexpr int BS = 72;  // B LDS row stride in dwords (64 data + 8 pad)
  __shared__ float sA[2][128 * AS];
  __shared__ float sB[2][KC * BS];  // BT=true uses it as [64][AS] (2176 <= 2304)

  const int lane = threadIdx.x & 31;
  const int wId  = threadIdx.x >> 5;
  const int half = lane >> 4;
  const int l15  = lane & 15;

  const int nStr  = Nc >> 6;
  const int mBlk  = (int)blockIdx.x / nStr;
  const int n0    = ((int)blockIdx.x % nStr) << 6;
  const int mBase = mBlk << 7;
  int rows = M - mBase; if (rows > 128) rows = 128;
  const int  m0  = mBase + (wId << 4);
  const bool act = (m0 < M);

  const int nChunks = K / KC;

  auto issue_chunk = [&](int c, int nb) {
    const int kc = c * KC;
    // A panel: rows x KC, row stride K
    tdm_load_2d((unsigned)(uintptr_t)&sA[nb][0],
                A + (size_t)mBase * K + kc,
                (unsigned)K, (unsigned)rows, (unsigned long long)K,
                KC, (unsigned)rows, /*2<<4=32 dw*/ 4u, /*2 dw pad*/ 1u);
    if (BT) {
      // Bt panel: 64 x KC, row stride K (no transpose needed: lanes read along k)
      tdm_load_2d((unsigned)(uintptr_t)&sB[nb][0],
                  B + (size_t)n0 * K + kc,
                  (unsigned)K, 64u, (unsigned long long)K,
                  KC, 64u, 4u, 1u);
    } else {
      // B panel: KC x 64, row stride Nc
      tdm_load_2d((unsigned)(uintptr_t)&sB[nb][0],
                  B + (size_t)kc * Nc + n0,
                  (unsigned)Nc, KC, (unsigned long long)Nc,
                  64u, KC, /*2<<5=64 dw*/ 5u, /*8 dw pad*/ 7u);
    }
  };

  if (threadIdx.x < 32) issue_chunk(0, 0);

  v8f acc0 = {}, acc1 = {}, acc2 = {}, acc3 = {};

  for (int c = 0; c < nChunks; ++c) {
    const int cb = c & 1;
    if (c + 1 < nChunks) {
      if (threadIdx.x < 32) issue_chunk(c + 1, cb ^ 1);
      __builtin_amdgcn_s_wait_tensorcnt((short)2);  // chunk c's pair has landed
    } else {
      __builtin_amdgcn_s_wait_tensorcnt((short)0);
    }
    __syncthreads();  // LDS chunk visible to all waves

    if (act) {
      const float* aC = &sA[cb][0];
      const float* bC = &sB[cb][0];
#pragma unroll
      for (int kk = 0; kk < KC; kk += 4) {
        // A 16x4 fragment: VGPR0 = K=kk (lanes<16) / kk+2 ; VGPR1 = kk+1 / kk+3
        v2f a = *(const v2f*)(aC + (size_t)((wId << 4) + l15) * AS + kk + (half << 1));
        v8f* accs[4] = {&acc0, &acc1, &acc2, &acc3};
#pragma unroll
        for (int t = 0; t < 4; ++t) {
          v2f b;
          if (BT) {
            b = *(const v2f*)(bC + (size_t)((t << 4) + l15) * AS + kk + (half << 1));
          } else {
            const int r = kk + (half << 1);
            b.x = bC[(size_t)r * BS + (t << 4) + l15];
            b.y = bC[(size_t)(r + 1) * BS + (t << 4) + l15];
          }
          *accs[t] = __builtin_amdgcn_wmma_f32_16x16x4_f32(
              false, a, false, b, (short)0, *accs[t], false, false);
        }
      }
    }
    __syncthreads();  // chunk cb fully consumed before it is overwritten
  }

  if (act) {
    v8f outs[4] = {acc0, acc1, acc2, acc3};
#pragma unroll
    for (int t = 0; t < 4; ++t) {
      const int col = n0 + (t << 4) + l15;
      const float bv = bias ? bias[col] : 0.0f;
#pragma unroll
      for (int i = 0; i < 8; ++i) {
        const int row = m0 + i + (half << 3);
        C[(size_t)row * Nc + col] = outs[t][i] + bv;
      }
    }
  }
}

// ---------------------------------------------------------------------------
// Attention logit precompute: als[n,h] = <h[n,h,:], a_src[h,:]>, ald likewise.
// ---------------------------------------------------------------------------
template <int H>
__global__ void attn_logits_kernel(const float* __restrict__ hmat,
                                   const float* __restrict__ asrc,
                                   const float* __restrict__ adst,
                                   float* __restrict__ als, float* __restrict__ ald,
                                   int Nn) {
  const int t = blockIdx.x * blockDim.x + threadIdx.x;
  if (t >= Nn * H) return;
  const int n = t / H;
  const int hh = t - n * H;
  const float* hp = hmat + (size_t)n * (H * 64) + hh * 64;
  const float* sp = asrc + hh * 64;
  const float* dp = adst + hh * 64;
  float s = 0.f, d = 0.f;
#pragma unroll
  for (int c = 0; c < 64; c += 4) {
    float4 hv = *(const float4*)(hp + c);
    float4 sv = *(const float4*)(sp + c);
    float4 dv = *(const float4*)(dp + c);
    s += hv.x * sv.x + hv.y * sv.y + hv.z * sv.z + hv.w * sv.w;
    d += hv.x * dv.x + hv.y * dv.y + hv.z * dv.z + hv.w * dv.w;
  }
  als[t] = s;
  ald[t] = d;
}

__global__ void init_stats_kernel(float* __restrict__ m, float* __restrict__ d, int n) {
  const int t = blockIdx.x * blockDim.x + threadIdx.x;
  if (t < n) { m[t] = -__builtin_inff(); d[t] = 0.0f; }
}

__global__ void zero_kernel(float* __restrict__ p, long long n) {
  const long long t = (long long)blockIdx.x * blockDim.x + threadIdx.x;
  if (t < n) p[t] = 0.0f;
}

__device__ inline void atomicMaxF(float* addr, float val) {
  if (val >= 0.0f) atomicMax((int*)addr, __float_as_int(val));
  else             atomicMin((unsigned int*)addr, __float_as_uint(val));
}

// ---------------------------------------------------------------------------
// Segment softmax pass 1: running max of leaky_relu(als[src]+ald[dst]) per dst.
// ---------------------------------------------------------------------------
template <int H>
__global__ void edge_max_kernel(const int* __restrict__ ei, int E, int Nn,
                                const float* __restrict__ als,
                                const float* __restrict__ ald,
                                float* __restrict__ mbuf) {
  const int t = blockIdx.x * blockDim.x + threadIdx.x;
  const int Etot = E + Nn;
  if (t >= Etot * H) return;
  const int e = t / H;
  const int hh = t - e * H;
  const int src = (e < E) ? ei[e] : (e - E);
  const int dst = (e < E) ? ei[E + e] : (e - E);
  float v = als[src * H + hh] + ald[dst * H + hh];
  v = (v > 0.0f) ? v : 0.2f * v;
  atomicMaxF(&mbuf[dst * H + hh], v);
}

// Segment softmax pass 2: denom = sum exp(e - m[dst]).
template <int H>
__global__ void edge_denom_kernel(const int* __restrict__ ei, int E, int Nn,
                                  const float* __restrict__ als,
                                  const float* __restrict__ ald,
                                  const float* __restrict__ mbuf,
                                  float* __restrict__ dbuf) {
  const int t = blockIdx.x * blockDim.x + threadIdx.x;
  const int Etot = E + Nn;
  if (t >= Etot * H) return;
  const int e = t / H;
  const int hh = t - e * H;
  const int src = (e < E) ? ei[e] : (e - E);
  const int dst = (e < E) ? ei[E + e] : (e - E);
  float v = als[src * H + hh] + ald[dst * H + hh];
  v = (v > 0.0f) ? v : 0.2f * v;
  atomicAdd(&dbuf[dst * H + hh], __expf(v - mbuf[dst * H + hh]));
}

// Pass 3: out[dst] += alpha * h[src].  One wave per edge; alpha recomputed
// uniformly across lanes (broadcast loads), payload scattered with f32 atomics
// (all operands resident in the 192MB L2).
template <int H>
__global__ __launch_bounds__(256) void edge_agg_kernel(
    const int* __restrict__ ei, int E, int Nn,
    const float* __restrict__ hmat,
    const float* __restrict__ als, const float* __restrict__ ald,
    const float* __restrict__ mbuf, const float* __restrict__ dbuf,
    float* __restrict__ outv) {
  const int lane = threadIdx.x & 31;
  const int e = blockIdx.x * (blockDim.x >> 5) + (threadIdx.x >> 5);
  const int Etot = E + Nn;
  if (e >= Etot) return;
  const int src = (e < E) ? ei[e] : (e - E);
  const int dst = (e < E) ? ei[E + e] : (e - E);

  float alpha[H];
#pragma unroll
  for (int hh = 0; hh < H; ++hh) {
    float v = als[src * H + hh] + ald[dst * H + hh];
    v = (v > 0.0f) ? v : 0.2f * v;
    const float ex = __expf(v - mbuf[dst * H + hh]);
    alpha[hh] = ex / (dbuf[dst * H + hh] + 1e-16f);
  }

  const float* hp = hmat + (size_t)src * (H * 64);
  float* op = outv + (size_t)dst * (H * 64);
#pragma unroll
  for (int i = 0; i < H * 2; ++i) {
    const int idx = i * 32 + lane;
    atomicAdd(&op[idx], hp[idx] * alpha[i >> 1]);
  }
}

__global__ void bias_act_kernel(float* __restrict__ v, const float* __restrict__ bias,
                                long long total, int mask, int elu) {
  const long long t = (long long)blockIdx.x * blockDim.x + threadIdx.x;
  if (t >= total) return;
  float x = v[t] + bias[(int)t & mask];
  if (elu) x = (x > 0.0f) ? x : (__expf(x) - 1.0f);
  v[t] = x;
}

// ---------------------------------------------------------------------------
// Final heads: one wave per node; 5 dot products of length 64 via shuffles.
// out layout: [N,3] logits, then [N] cpu sigmoid, then [N] mem sigmoid.
// ---------------------------------------------------------------------------
__global__ __launch_bounds__(256) void heads_kernel(
    const float* __restrict__ a,
    const float* __restrict__ Wlin, const float* __restrict__ blin,
    const float* __restrict__ Wcpu, const float* __restrict__ bcpu,
    const float* __restrict__ Wmem, const float* __restrict__ bmem,
    float* __restrict__ out, int Nn) {
  const int lane = threadIdx.x & 31;
  const int n = blockIdx.x * (blockDim.x >> 5) + (threadIdx.x >> 5);
  if (n >= Nn) return;
  const float2 av = *(const float2*)(a + (size_t)n * 64 + lane * 2);
  const float* Ws[5] = {Wlin, Wlin + 64, Wlin + 128, Wcpu, Wmem};
  float r[5];
#pragma unroll
  for (int j = 0; j < 5; ++j) {
    const float2 wv = *(const float2*)(Ws[j] + lane * 2);
    float p = av.x * wv.x + av.y * wv.y;
#pragma unroll
    for (int s = 16; s > 0; s >>= 1) p += __shfl_xor(p, s, 32);
    r[j] = p;
  }
  if (lane == 0) {
    out[(size_t)n * 3 + 0] = r[0] + blin[0];
    out[(size_t)n * 3 + 1] = r[1] + blin[1];
    out[(size_t)n * 3 + 2] = r[2] + blin[2];
    out[(size_t)3 * Nn + n] = 1.0f / (1.0f + __expf(-(r[3] + bcpu[0])));
    out[(size_t)4 * Nn + n] = 1.0f / (1.0f + __expf(-(r[4] + bmem[0])));
  }
}

// ---------------------------------------------------------------------------
// Host-side helpers
// ---------------------------------------------------------------------------
static void launch_gemm(const float* A, const float* B, const float* bias, float* C,
                        int M, int K, int Nc, bool bt, hipStream_t stream) {
  const int blocks = cdiv_i(M, 128) * (Nc / 64);
  if (bt) gemm_wmma_kernel<true><<<blocks, 256, 0, stream>>>(A, B, bias, C, M, K, Nc);
  else    gemm_wmma_kernel<false><<<blocks, 256, 0, stream>>>(A, B, bias, C, M, K, Nc);
}

template <int H>
static void gat_aggregate(const int* ei, int E, int Nn, const float* hmat,
                          const float* asrc, const float* adst,
                          float* als, float* ald, float* mbuf, float* dbuf,
                          float* outv, const float* bias, int elu, hipStream_t stream) {
  const int F = H * 64;
  const int Etot = E + Nn;
  attn_logits_kernel<H><<<cdiv_i((long long)Nn * H, 256), 256, 0, stream>>>(
      hmat, asrc, adst, als, ald, Nn);
  init_stats_kernel<<<cdiv_i((long long)Nn * H, 256), 256, 0, stream>>>(mbuf, dbuf, Nn * H);
  zero_kernel<<<cdiv_i((long long)Nn * F, 256), 256, 0, stream>>>(outv, (long long)Nn * F);
  edge_max_kernel<H><<<cdiv_i((long long)Etot * H, 256), 256, 0, stream>>>(
      ei, E, Nn, als, ald, mbuf);
  edge_denom_kernel<H><<<cdiv_i((long long)Etot * H, 256), 256, 0, stream>>>(
      ei, E, Nn, als, ald, mbuf, dbuf);
  edge_agg_kernel<H><<<cdiv_i(Etot, 8), 256, 0, stream>>>(
      ei, E, Nn, hmat, als, ald, mbuf, dbuf, outv);
  bias_act_kernel<<<cdiv_i((long long)Nn * F, 256), 256, 0, stream>>>(
      outv, bias, (long long)Nn * F, F - 1, elu);
}

extern "C" void kernel_launch(void* const* d_in, const int* in_sizes, int n_in,
                              void* d_out, int out_size, void* d_ws, size_t ws_size,
                              hipStream_t stream) {
  const float* x    = (const float*)d_in[0];
  const int*   ei   = (const int*)d_in[1];
  const float* W1   = (const float*)d_in[2];
  const float* as1  = (const float*)d_in[3];
  const float* ad1  = (const float*)d_in[4];
  const float* b1   = (const float*)d_in[5];
  const float* W2   = (const float*)d_in[6];
  const float* as2  = (const float*)d_in[7];
  const float* ad2  = (const float*)d_in[8];
  const float* b2   = (const float*)d_in[9];
  const float* W3   = (const float*)d_in[10];
  const float* as3  = (const float*)d_in[11];
  const float* ad3  = (const float*)d_in[12];
  const float* b3   = (const float*)d_in[13];
  const float* Wqkv = (const float*)d_in[14];
  const float* bqkv = (const float*)d_in[15];
  const float* Wo   = (const float*)d_in[16];
  const float* bo   = (const float*)d_in[17];
  const float* Wlin = (const float*)d_in[18];
  const float* blin = (const float*)d_in[19];
  const float* Wcpu = (const float*)d_in[20];
  const float* bcpu = (const float*)d_in[21];
  const float* Wmem = (const float*)d_in[22];
  const float* bmem = (const float*)d_in[23];

  const int Nn = in_sizes[0] / 256;
  const int E  = in_sizes[1] / 2;

  float* ws   = (float*)d_ws;
  float* bufA = ws;                               // N*256 floats
  float* bufB = bufA + (size_t)Nn * 256;          // N*256 floats
  float* als  = bufB + (size_t)Nn * 256;          // N*4
  float* ald  = als + (size_t)Nn * 4;             // N*4
  float* mbuf = ald + (size_t)Nn * 4;             // N*4
  float* dbuf = mbuf + (size_t)Nn * 4;            // N*4

  // conv1: 256 -> 4 heads x 64
  launch_gemm(x, W1, nullptr, bufA, Nn, 256, 256, false, stream);
  gat_aggregate<4>(ei, E, Nn, bufA, as1, ad1, als, ald, mbuf, dbuf, bufB, b1, 1, stream);

  // conv2: 256 -> 2 heads x 64
  launch_gemm(bufB, W2, nullptr, bufA, Nn, 256, 128, false, stream);
  gat_aggregate<2>(ei, E, Nn, bufA, as2, ad2, als, ald, mbuf, dbuf, bufB, b2, 1, stream);

  // conv3: 128 -> 1 head x 64 (no ELU)
  launch_gemm(bufB, W3, nullptr, bufA, Nn, 128, 64, false, stream);
  gat_aggregate<1>(ei, E, Nn, bufA, as3, ad3, als, ald, mbuf, dbuf, bufB, b3, 0, stream);

  // seq_len==1 MHA: softmax over one key == 1  =>  ctx = V = h @ Wv^T + bv
  launch_gemm(bufB, Wqkv + 128 * 64, bqkv + 128, bufA, Nn, 64, 64, true, stream);
  // a = ctx @ Wo^T + bo
  launch_gemm(bufA, Wo, bo, bufB, Nn, 64, 64, true, stream);

  // heads -> d_out
  heads_kernel<<<cdiv_i(Nn, 8), 256, 0, stream>>>(
      bufB, Wlin, blin, Wcpu, bcpu, Wmem, bmem, (float*)d_out, Nn);
}